// HBVMul_73297911874035
// MI455X (gfx1250) — compile-verified
//
#include <hip/hip_runtime.h>
#include <hip/hip_bf16.h>
#include <stdint.h>
#include <math.h>

#define T_STEPS 730
#define NGRID   4000
#define MU      16
#define GPB     16              // grids per block
#define BLOCK   256             // 16 grids * 16 mu = 8 waves
#define CT      16              // timesteps per LDS chunk
#define CH_ELEMS (CT * GPB * 3) // 768 floats per chunk
#define ISSUES_PER_THREAD (CH_ELEMS / BLOCK) // 3

// ---------------- CDNA5 async global->LDS path (guarded) ----------------
#if __has_builtin(__builtin_amdgcn_global_load_async_to_lds_b32) && \
    __has_builtin(__builtin_amdgcn_s_wait_asynccnt)
#define HAS_ASYNC_LDS 1
#else
#define HAS_ASYNC_LDS 0
#endif

__device__ __forceinline__ void async_copy_f32(const float* g, float* l) {
#if HAS_ASYNC_LDS
  // builtin params are int* in addrspace(1)/(3); generic LDS pointer's low
  // 32 bits are the LDS byte address (ISA aperture rule)
  __builtin_amdgcn_global_load_async_to_lds_b32(
      (__attribute__((address_space(1))) int*)(uintptr_t)g,
      (__attribute__((address_space(3))) int*)(uint32_t)(uintptr_t)l,
      0, 0);
#else
  *l = *g;  // fallback: plain vmem load + ds write
#endif
}

template <int N>
__device__ __forceinline__ void wait_async() {
#if HAS_ASYNC_LDS
  __builtin_amdgcn_s_wait_asynccnt(N);
#endif
}

// fast transcendentals: v_log_f32 (log2) and v_exp_f32 (2^x)
__device__ __forceinline__ float fast_log2(float v) {
#if __has_builtin(__builtin_amdgcn_logf)
  return __builtin_amdgcn_logf(v);
#else
  return log2f(v);
#endif
}
__device__ __forceinline__ float fast_exp2(float v) {
#if __has_builtin(__builtin_amdgcn_exp2f)
  return __builtin_amdgcn_exp2f(v);
#else
  return exp2f(v);
#endif
}

// Stage one chunk of forcing data x[t0 .. t0+CT) for this block's 16 grids.
// 768 contiguous floats in LDS; each of 256 threads issues exactly 3 copies
// (uniform per wave => ASYNCcnt bookkeeping is exact). Tail clamps t (redundant
// loads, never read) so the issue count stays uniform.
__device__ __forceinline__ void prefetch_chunk(const float* __restrict__ x,
                                               float* __restrict__ dst,
                                               int t0, int gBase) {
#pragma unroll
  for (int r = 0; r < ISSUES_PER_THREAD; ++r) {
    int e  = (int)threadIdx.x + r * BLOCK;  // 0..767
    int tt = e / 48;                        // local timestep
    int j  = e - tt * 48;                   // grid*3 + channel within tile
    int tl = t0 + tt;
    if (tl > T_STEPS - 1) tl = T_STEPS - 1; // clamp, keep EXEC full & count uniform
    const float* g = x + (size_t)tl * (NGRID * 3) + (size_t)gBase * 3 + j;
    async_copy_f32(g, dst + e);
  }
}

__global__ void __launch_bounds__(BLOCK)
hbv_kernel(const float* __restrict__ x,        // (730, 4000, 3)
           const float* __restrict__ par,      // (4000, 14, 16)
           float* __restrict__ out) {          // (730, 4000)
  __shared__ float buf[2][CH_ELEMS];

  const int tid   = threadIdx.x;
  const int mu    = tid & 15;   // ensemble member
  const int gl    = tid >> 4;   // local grid 0..15
  const int gBase = blockIdx.x * GPB;
  const int g     = gBase + gl;

  // hint: pull this block's parameter slab toward the caches (global_prefetch_b8)
  __builtin_prefetch(par + (size_t)gBase * (14 * 16), 0, 1);

  // ---- parameter load + bound scaling (done once) ----
  const float lo[12] = {1.f, 50.f,   0.05f, 0.01f, 0.001f, 0.2f, 0.f,  0.f,   -2.5f, 0.5f, 0.f,  0.f };
  const float hi[12] = {6.f, 1000.f, 0.9f,  0.5f,  0.2f,   1.f,  10.f, 100.f,  2.5f, 10.f, 0.1f, 0.2f};
  float p[12];
  const float* pb = par + (size_t)g * (14 * 16) + mu;
#pragma unroll
  for (int i = 0; i < 12; ++i) p[i] = lo[i] + pb[i * 16] * (hi[i] - lo[i]);

  const float parBETA = p[0], parFC  = p[1], parK0   = p[2],  parK1  = p[3];
  const float parK2   = p[4], parLP  = p[5], parPERC = p[6],  parUZL = p[7];
  const float parTT   = p[8], parCFMAX = p[9], parCFR = p[10], parCWH = p[11];

  const float invFC   = 1.0f / parFC;
  const float invLPFC = 1.0f / (parLP * parFC);
  const float cfrcf   = parCFR * parCFMAX;

  // ---- state (VGPR-resident across all 730 steps) ----
  float SP = 0.001f, MW = 0.001f, SM = 0.001f, SUZ = 0.001f, SLZ = 0.001f;

  prefetch_chunk(x, &buf[0][0], 0, gBase);

  const int nch = (T_STEPS + CT - 1) / CT;  // 46
  for (int k = 0; k < nch; ++k) {
    if (k + 1 < nch) {
      prefetch_chunk(x, &buf[(k + 1) & 1][0], (k + 1) * CT, gBase);
      wait_async<ISSUES_PER_THREAD>();   // retire chunk k, leave k+1 in flight
    } else {
      wait_async<0>();
    }
    __syncthreads();

    const float* cb = &buf[k & 1][0] + gl * 3;
    const int t0   = k * CT;
    const int tcnt = (T_STEPS - t0 < CT) ? (T_STEPS - t0) : CT;

    for (int ts = 0; ts < tcnt; ++ts) {
      const float P  = cb[ts * 48 + 0];
      const float Tt = cb[ts * 48 + 1];
      const float E  = cb[ts * 48 + 2];

      const bool  cold = (Tt < parTT);
      const float RAIN = cold ? 0.0f : P;
      const float SNOW = cold ? P : 0.0f;

      SP += SNOW;
      const float melt = fminf(fmaxf(parCFMAX * (Tt - parTT), 0.0f), SP);
      MW += melt; SP -= melt;
      const float refr = fminf(fmaxf(cfrcf * (parTT - Tt), 0.0f), MW);
      SP += refr; MW -= refr;
      const float tosoil = fmaxf(MW - parCWH * SP, 0.0f);
      MW -= tosoil;

      // soil wetness = clip((SM/FC)^BETA, 0, 1); SM in (0, FC] so log2 is finite
      const float r  = SM * invFC;
      const float sw = fminf(fast_exp2(parBETA * fast_log2(r)), 1.0f);
      const float rech = (RAIN + tosoil) * sw;
      SM += RAIN + tosoil - rech;
      const float excess = fmaxf(SM - parFC, 0.0f);
      SM -= excess;
      const float evapf = fminf(fmaxf(SM * invLPFC, 0.0f), 1.0f);
      const float ETa   = fminf(SM, E * evapf);
      SM = fmaxf(SM - ETa, 1e-5f);

      SUZ += rech + excess;
      const float PERC = fminf(SUZ, parPERC);
      SUZ -= PERC;
      const float Q0 = parK0 * fmaxf(SUZ - parUZL, 0.0f);
      SUZ -= Q0;
      const float Q1 = parK1 * SUZ;
      SUZ -= Q1;
      SLZ += PERC;
      const float Q2 = parK2 * SLZ;
      SLZ -= Q2;

      // ensemble mean over 16 mu lanes (butterfly within each 16-lane half-wave)
      float q = Q0 + Q1 + Q2;
      q += __shfl_xor(q, 1, 32);
      q += __shfl_xor(q, 2, 32);
      q += __shfl_xor(q, 4, 32);
      q += __shfl_xor(q, 8, 32);
      if (mu == 0) out[(size_t)(t0 + ts) * NGRID + g] = q * (1.0f / 16.0f);
    }
    __syncthreads();  // protect buf[k&1] before it is overwritten at k+2
  }
}

extern "C" void kernel_launch(void* const* d_in, const int* in_sizes, int n_in,
                              void* d_out, int out_size, void* d_ws, size_t ws_size,
                              hipStream_t stream) {
  const float* x   = (const float*)d_in[0];   // (730, 4000, 3) f32
  const float* par = (const float*)d_in[1];   // (4000, 14, 16) f32
  float* out = (float*)d_out;                 // (730, 4000, 1) f32
  (void)in_sizes; (void)n_in; (void)d_ws; (void)ws_size; (void)out_size;

  dim3 grid(NGRID / GPB);  // 250 blocks
  dim3 block(BLOCK);       // 256 threads = 8 wave32
  hbv_kernel<<<grid, block, 0, stream>>>(x, par, out);
}